// FDLT_70592082477318
// MI455X (gfx1250) — compile-verified
//
#include <hip/hip_runtime.h>

// FDLT fused kernels for MI455X (gfx1250, wave32, WMMA, async-to-LDS DMA).
//
// Reference:
//   psi = psiHat * cm
//   psi[:,0::2,:] @= XFc ; psi[:,1::2,:] @= XFs        (256x256 each)
//   out[b,m,o] = sum_i psi[b,m,i] * D[m,o,i]
//
// Fusion: out[b,m,:] = cm * psiHat[b,m,:] @ E[m], E[m] = XF_p(m) @ D[m]^T.
// Stored transposed: Et[m][o][j] = sum_i D[m][o][i] * XF_p[j][i]
//   -> both passes are C = A . Bt^T with K-contiguous operands (K=256).

typedef __attribute__((ext_vector_type(2))) float v2f;
typedef __attribute__((ext_vector_type(8))) float v8f;

#define BV    128
#define NN    256          // 2*B = K dimension everywhere
#define KB    32           // K block staged in LDS
#define MT    64           // C row tile per workgroup
#define NT    128          // C col tile per workgroup
#define BUF_FLOATS (MT*KB + NT*KB)   // 2048 + 4096 = 6144 floats = 24 KB
#define SB_OFF_B   (MT*KB*4)         // byte offset of Bt region in a buffer

// One 16-byte async global->LDS copy (ASYNCcnt-tracked, CDNA5 path).
__device__ __forceinline__ void async_copy16(unsigned lds_byte, const float* gptr) {
    asm volatile("global_load_async_to_lds_b128 %0, %1, off"
                 :: "v"(lds_byte), "v"(gptr)
                 : "memory");
}
__device__ __forceinline__ void wait_async0() {
    asm volatile("s_wait_asynccnt 0x0" ::: "memory");
}

// Stage one K-block (A: MTxKB, Bt: NTxKB) into LDS buffer via async DMA.
// 256 threads; each issues 2 A-loads + 4 B-loads of 16B.
__device__ __forceinline__ void stage_async(
    const float* __restrict__ A,  long lda,
    const float* __restrict__ Bt, long ldb,
    int kb, unsigned bufBaseByte, int tid)
{
    #pragma unroll
    for (int s = 0; s < 2; ++s) {               // A: 64 rows x 32 floats
        int t   = tid + s * 256;
        int row = t >> 3;                        // 0..63
        int c4  = (t & 7) << 2;                  // 0..28
        async_copy16(bufBaseByte + (unsigned)(row * KB + c4) * 4u,
                     A + (long)row * lda + kb + c4);
    }
    #pragma unroll
    for (int s = 0; s < 4; ++s) {               // Bt: 128 rows x 32 floats
        int t   = tid + s * 256;
        int row = t >> 3;                        // 0..127
        int c4  = (t & 7) << 2;
        async_copy16(bufBaseByte + SB_OFF_B + (unsigned)(row * KB + c4) * 4u,
                     Bt + (long)row * ldb + kb + c4);
    }
}

// C(64x128) = A(64x256) * Bt(128x256)^T, double-buffered LDS + WMMA f32.
// 256 threads = 8 waves; wave w -> rows (w&1)*32 + {0,16}, cols (w>>1)*32 + {0,16}.
__device__ __forceinline__ void gemm_abt_64x128(
    const float* __restrict__ A,  long lda,
    const float* __restrict__ Bt, long ldb,
    float* __restrict__ C,        long ldc,
    float alpha)
{
    __shared__ __align__(16) float smem[2 * BUF_FLOATS];   // 48 KB
    const unsigned ldsBase = (unsigned)(size_t)&smem[0];   // LDS aperture: offset in addr[31:0]

    const int tid  = threadIdx.x;
    const int lane = tid & 31;
    const int wave = tid >> 5;
    const int half = lane >> 4;
    const int l16  = lane & 15;
    const int rt   = (wave & 1) * 32;        // row base (two 16-row tiles)
    const int ct   = (wave >> 1) * 32;       // col base (two 16-col tiles)

    v8f acc00 = {}, acc01 = {}, acc10 = {}, acc11 = {};

    // Prologue: fill buffer 0 with K-block 0.
    stage_async(A, lda, Bt, ldb, 0, 0u + ldsBase, tid);
    wait_async0();
    __syncthreads();

    for (int kb = 0; kb < NN / KB; ++kb) {
        const int cur = kb & 1;
        if (kb + 1 < NN / KB)   // uniform branch: EXEC stays all-ones
            stage_async(A, lda, Bt, ldb, (kb + 1) * KB,
                        ldsBase + (unsigned)(1 - cur) * (BUF_FLOATS * 4), tid);

        const float* sA = &smem[cur * BUF_FLOATS];
        const float* sB = sA + MT * KB;

        #pragma unroll
        for (int ks = 0; ks < KB / 4; ++ks) {
            // A frags 16x4: lane l16 = M row, half selects K pair {0,1}/{2,3}.
            v2f a0 = *(const v2f*)&sA[(rt      + l16) * KB + ks * 4 + half * 2];
            v2f a1 = *(const v2f*)&sA[(rt + 16 + l16) * KB + ks * 4 + half * 2];
            // B frags 4x16: lane l16 = N col.
            v2f b0 = *(const v2f*)&sB[(ct      + l16) * KB + ks * 4 + half * 2];
            v2f b1 = *(const v2f*)&sB[(ct + 16 + l16) * KB + ks * 4 + half * 2];
            acc00 = __builtin_amdgcn_wmma_f32_16x16x4_f32(false, a0, false, b0, (short)0, acc00, false, false);
            acc01 = __builtin_amdgcn_wmma_f32_16x16x4_f32(false, a0, false, b1, (short)0, acc01, false, false);
            acc10 = __builtin_amdgcn_wmma_f32_16x16x4_f32(false, a1, false, b0, (short)0, acc10, false, false);
            acc11 = __builtin_amdgcn_wmma_f32_16x16x4_f32(false, a1, false, b1, (short)0, acc11, false, false);
        }

        wait_async0();        // next buffer's DMA (this wave's) complete
        __syncthreads();      // all waves: writes visible, current buffer free
    }

    // C/D layout: VGPR v -> M = v + 8*half, N = l16.
    #pragma unroll
    for (int v = 0; v < 8; ++v) {
        int r0 = rt + v + half * 8;
        int r1 = r0 + 16;
        C[(long)r0 * ldc + ct      + l16] = alpha * acc00[v];
        C[(long)r0 * ldc + ct + 16 + l16] = alpha * acc01[v];
        C[(long)r1 * ldc + ct      + l16] = alpha * acc10[v];
        C[(long)r1 * ldc + ct + 16 + l16] = alpha * acc11[v];
    }
}

// Kernel 1: Et[m][o][j] = sum_i D[m][o][i] * XF_p[j][i]
// grid: x = oc(2)*jc(2) -> 4, y = m(128)
__global__ __launch_bounds__(256) void fdlt_build_Et(
    const float* __restrict__ D,
    const float* __restrict__ XFc,
    const float* __restrict__ XFs,
    float* __restrict__ Et)
{
    const int m  = blockIdx.y;
    const int oc = blockIdx.x >> 1;           // 0..1  (o chunk of 64)
    const int jc = blockIdx.x & 1;            // 0..1  (j chunk of 128)
    const float* XF = (m & 1) ? XFs : XFc;

    const float* A  = D  + (size_t)m * (BV * NN) + (size_t)(oc * MT) * NN;
    const float* Bt = XF + (size_t)(jc * NT) * NN;
    float*       C  = Et + (size_t)m * (BV * NN) + (size_t)(oc * MT) * NN + jc * NT;

    gemm_abt_64x128(A, NN, Bt, NN, C, NN, 1.0f);
}

// Kernel 2: out[b][m][o] = cm * sum_j psiHat[b][m][j] * Et[m][o][j]
// grid: x = batch tile (32 tiles of 64), y = m(128)
__global__ __launch_bounds__(256) void fdlt_main(
    const float* __restrict__ psiHat,
    const float* __restrict__ Et,
    const float* __restrict__ cm_p,
    float* __restrict__ out)
{
    const int m  = blockIdx.y;
    const int bt = blockIdx.x;
    const float cm = *cm_p;

    const float* A  = psiHat + (size_t)(bt * MT) * (BV * NN) + (size_t)m * NN;
    const float* Bt = Et     + (size_t)m * (BV * NN);
    float*       C  = out    + (size_t)(bt * MT) * (BV * BV) + (size_t)m * BV;

    gemm_abt_64x128(A, (long)BV * NN, Bt, NN, C, (long)BV * BV, cm);
}

extern "C" void kernel_launch(void* const* d_in, const int* in_sizes, int n_in,
                              void* d_out, int out_size, void* d_ws, size_t ws_size,
                              hipStream_t stream) {
    const float* psiHat = (const float*)d_in[0];
    const float* cm     = (const float*)d_in[1];
    const float* XFc    = (const float*)d_in[2];
    const float* XFs    = (const float*)d_in[3];
    const float* D      = (const float*)d_in[4];
    float*       out    = (float*)d_out;
    float*       Et     = (float*)d_ws;   // 128*256*128 floats = 16.8 MB

    hipLaunchKernelGGL(fdlt_build_Et, dim3(4, 128), dim3(256), 0, stream,
                       D, XFc, XFs, Et);
    hipLaunchKernelGGL(fdlt_main, dim3(32, 128), dim3(256), 0, stream,
                       psiHat, Et, cm, out);
}